// ProcessingLayer_33328946217352
// MI455X (gfx1250) — compile-verified
//
#include <hip/hip_runtime.h>
#include <math.h>

typedef float v2f __attribute__((ext_vector_type(2)));
typedef float v8f __attribute__((ext_vector_type(8)));

#define BB   32
#define NPTT 1024

__device__ __forceinline__ float eluf(float x) {
    return x > 0.f ? x : (__expf(x) - 1.f);
}

// ---------------------------------------------------------------------------
// Generic fp32 GEMM via V_WMMA_F32_16X16X4_F32. One wave32 per 16x32 C tile
// (two 16x16 WMMA accumulators sharing the A fragment).
//   A: rows mapped m -> (m / rowsPerBatch)*batchStrideA + (m % rowsPerBatch)*lda
//   B: row-major Kdim x N (ldb == N)
//   Out: row-major, row stride ldo (pointer may be pre-offset for column blocks)
//   act: 0 = none, 1 = ELU, 2 = ELU then BN(g,b), 3 = BN only
// All out-of-range loads are done branchlessly: addresses are clamped into the
// valid region and the loaded value is replaced by 0 via select (v_cndmask),
// so the K-loop carries no EXEC-mask branches around the WMMAs.
// ---------------------------------------------------------------------------
__global__ __launch_bounds__(256)
void gemm_wmma_f32(const float* __restrict__ A, int lda, int rowsPerBatch,
                   long long batchStrideA,
                   const float* __restrict__ Bm,
                   float* __restrict__ Out, int ldo,
                   const float* __restrict__ gamma, const float* __restrict__ beta,
                   int M, int Kdim, int N, int act)
{
    const int ntilesN = (N + 31) >> 5;           // 32-wide tiles
    const int ntilesM = (M + 15) >> 4;
    const int tile = blockIdx.x * blockDim.y + threadIdx.y;
    if (tile >= ntilesM * ntilesN) return;       // wave-uniform exit

    const int tm = (tile / ntilesN) << 4;
    const int tn = (tile % ntilesN) << 5;

    const int lane = threadIdx.x;                // 0..31 (wave32)
    const int half = lane >> 4;                  // K-pair selector
    const int l16  = lane & 15;

    const int  mrow   = tm + l16;
    const bool mvalid = (mrow < M);
    const int  mr     = mvalid ? mrow : 0;       // clamped row
    const float* Arow = A + (long long)(mr / rowsPerBatch) * batchStrideA
                          + (long long)(mr % rowsPerBatch) * lda;

    const int  ncol0  = tn + l16;
    const int  ncol1  = tn + 16 + l16;
    const bool nv0    = (ncol0 < N);
    const bool nv1    = (ncol1 < N);
    const int  nc0    = nv0 ? ncol0 : (N - 1);   // clamped cols
    const int  nc1    = nv1 ? ncol1 : (N - 1);

    v8f acc0 = {0.f, 0.f, 0.f, 0.f, 0.f, 0.f, 0.f, 0.f};
    v8f acc1 = {0.f, 0.f, 0.f, 0.f, 0.f, 0.f, 0.f, 0.f};

    for (int kk = 0; kk < Kdim; kk += 4) {
        if ((kk & 15) == 0) {
            // wave-uniform speculative prefetch of upcoming B panel
            __builtin_prefetch(Bm + (long long)(kk + 16) * N + nc0, 0, 1);
        }
        const int  k0  = kk + half * 2;
        const bool kv0 = (k0 < Kdim);
        const bool kv1 = (k0 + 1 < Kdim);
        const int  k0c = kv0 ? k0 : (Kdim - 1);      // clamped K indices
        const int  k1c = kv1 ? (k0 + 1) : (Kdim - 1);

        // unconditional loads from clamped addresses, then branchless select
        const float a0r = Arow[k0c];
        const float a1r = Arow[k1c];
        const float b00 = Bm[(long long)k0c * N + nc0];
        const float b01 = Bm[(long long)k1c * N + nc0];
        const float b10 = Bm[(long long)k0c * N + nc1];
        const float b11 = Bm[(long long)k1c * N + nc1];

        v2f a, b0, b1;
        a.x  = (mvalid && kv0) ? a0r : 0.f;
        a.y  = (mvalid && kv1) ? a1r : 0.f;
        b0.x = (nv0 && kv0) ? b00 : 0.f;
        b0.y = (nv0 && kv1) ? b01 : 0.f;
        b1.x = (nv1 && kv0) ? b10 : 0.f;
        b1.y = (nv1 && kv1) ? b11 : 0.f;

        acc0 = __builtin_amdgcn_wmma_f32_16x16x4_f32(false, a, false, b0,
                                                     (short)0, acc0, false, false);
        acc1 = __builtin_amdgcn_wmma_f32_16x16x4_f32(false, a, false, b1,
                                                     (short)0, acc1, false, false);
    }

    // C/D layout: VGPR r -> row tm + r + 8*half ; col = tile_n_base + l16
    const int orow0 = tm + half * 8;
    if (nv0) {
        float g = 1.f, bta = 0.f;
        if (act >= 2) { g = gamma[ncol0]; bta = beta[ncol0]; }
        #pragma unroll
        for (int r = 0; r < 8; ++r) {
            const int orow = orow0 + r;
            if (orow < M) {
                float v = acc0[r];
                if (act == 1 || act == 2) v = eluf(v);
                if (act >= 2)             v = v * g + bta;
                Out[(long long)orow * ldo + ncol0] = v;
            }
        }
    }
    if (nv1) {
        float g = 1.f, bta = 0.f;
        if (act >= 2) { g = gamma[ncol1]; bta = beta[ncol1]; }
        #pragma unroll
        for (int r = 0; r < 8; ++r) {
            const int orow = orow0 + r;
            if (orow < M) {
                float v = acc1[r];
                if (act == 1 || act == 2) v = eluf(v);
                if (act >= 2)             v = v * g + bta;
                Out[(long long)orow * ldo + ncol1] = v;
            }
        }
    }
}

// ---------------------------------------------------------------------------
// KNN: one thread per query point; maintains sorted (ascending d2) best-KD,
// stable on ties (earlier index wins) to match jax.lax.top_k(-d2).
// Writes idx[q*K + k] = best[k*D].
// ---------------------------------------------------------------------------
__global__ void k_knn(const float* __restrict__ points, int* __restrict__ idx,
                      int P, int Nprev, int K, int D, int KD)
{
    const int t = blockIdx.x * blockDim.x + threadIdx.x;
    if (t >= BB * P) return;
    const int n = t / P, p = t % P;
    const float* q = points + (long long)(n * NPTT + p) * 6;
    const float qx = q[0], qy = q[1], qz = q[2];

    float bd[48];
    int   bi[48];
    for (int j = 0; j < KD; ++j) { bd[j] = 3.0e38f; bi[j] = 0; }

    const float* base = points + (long long)n * NPTT * 6;
    for (int i = 0; i < Nprev; ++i) {
        const float dx = base[i * 6 + 0] - qx;
        const float dy = base[i * 6 + 1] - qy;
        const float dz = base[i * 6 + 2] - qz;
        const float d2 = dx * dx + dy * dy + dz * dz;
        if (d2 < bd[KD - 1]) {
            int pos = KD - 1;
            while (pos > 0 && bd[pos - 1] > d2) {
                bd[pos] = bd[pos - 1];
                bi[pos] = bi[pos - 1];
                --pos;
            }
            bd[pos] = d2;
            bi[pos] = i;
        }
    }
    for (int k = 0; k < K; ++k) idx[(long long)t * K + k] = bi[k * D];
}

// ---------------------------------------------------------------------------
// Gather: nn_local[n,p,k,:] = pts[idx] - qrs ; f_in[n,p,k, cpf + c] = fts_prev[idx][c]
// ---------------------------------------------------------------------------
__global__ void k_gather(const float* __restrict__ points, const float* __restrict__ ftsPrev,
                         const int* __restrict__ idx,
                         float* __restrict__ nnLocal, float* __restrict__ fIn,
                         int P, int K, int Cprev, int cin, int cpfv, int NprevF)
{
    const int t = blockIdx.x * blockDim.x + threadIdx.x;
    const int total = BB * P * K;
    if (t >= total) return;
    const int n   = t / (P * K);
    const int rem = t % (P * K);
    const int p   = rem / K;
    const int i   = idx[t];

    const float* q = points + (long long)(n * NPTT + p) * 6;
    const float* x = points + (long long)(n * NPTT + i) * 6;
    nnLocal[(long long)t * 3 + 0] = x[0] - q[0];
    nnLocal[(long long)t * 3 + 1] = x[1] - q[1];
    nnLocal[(long long)t * 3 + 2] = x[2] - q[2];

    const float* f = ftsPrev + (long long)(n * NprevF + i) * Cprev;
    float* dst = fIn + (long long)t * cin + cpfv;
    for (int c = 0; c < Cprev; ++c) dst[c] = f[c];
}

// ---------------------------------------------------------------------------
// Depthwise einsum 'npkc,kcm->npcm'
// ---------------------------------------------------------------------------
__global__ void k_depthwise(const float* __restrict__ x, const float* __restrict__ Wd,
                            float* __restrict__ out, int NP, int Kk, int Cc, int Mm)
{
    const int t = blockIdx.x * blockDim.x + threadIdx.x;
    const long long total = (long long)NP * Cc * Mm;
    if (t >= total) return;
    const int np  = t / (Cc * Mm);
    const int rem = t % (Cc * Mm);
    const int c   = rem / Mm;
    const int m   = rem % Mm;
    const float* xb = x + (long long)np * Kk * Cc;
    float s = 0.f;
    for (int k = 0; k < Kk; ++k)
        s += xb[k * Cc + c] * Wd[(long long)(k * Cc + c) * Mm + m];
    out[(long long)t] = s;
}

// ---------------------------------------------------------------------------
// Per-point fX = X (KxK) @ f_in (K x cin), in place on f_in through LDS.
// One block per point.
// ---------------------------------------------------------------------------
__global__ void k_fx(const float* __restrict__ X, float* __restrict__ fIn, int K, int cin)
{
    __shared__ float sF[3840];   // max 16*240
    __shared__ float sX[256];    // max 16*16
    const int bp = blockIdx.x;
    const int sz = K * cin;
    float* fp = fIn + (long long)bp * sz;
    for (int i = threadIdx.x; i < sz; i += blockDim.x)    sF[i] = fp[i];
    for (int i = threadIdx.x; i < K * K; i += blockDim.x) sX[i] = X[(long long)bp * K * K + i];
    __syncthreads();
    for (int i = threadIdx.x; i < sz; i += blockDim.x) {
        const int k = i / cin, c = i % cin;
        float s = 0.f;
        for (int l = 0; l < K; ++l) s += sX[k * K + l] * sF[l * cin + c];
        fp[i] = s;
    }
}

// mean over points: out[b, c] = mean_p x[b, p, c]
__global__ void k_mean(const float* __restrict__ x, float* __restrict__ out, int P, int C)
{
    const int t = blockIdx.x * blockDim.x + threadIdx.x;
    if (t >= BB * C) return;
    const int b = t / C, c = t % C;
    float s = 0.f;
    for (int p = 0; p < P; ++p) s += x[(long long)(b * P + p) * C + c];
    out[t] = s / (float)P;
}

// ---------------------------------------------------------------------------
// Host orchestration
// ---------------------------------------------------------------------------
struct LayerP {
    const float *aW1, *aW2, *abn1b, *abn1g, *abn2b, *abn2g;
    const float *Wd1, *Wd2, *Wp1, *Wp2, *Wx0;
    const float *bnx0b, *bnx0g, *bnx1b, *bnx1g, *bnx2b, *bnx2g;
    const float *cWd, *cWp, *cbnb, *cbng;
    const float *dW1, *dW2, *dbn1b, *dbn1g, *dbn2b, *dbn2g;
};

static inline void launch_gemm(hipStream_t s, const float* A, int lda, int rpb, long long bstride,
                               const float* Bm, float* Out, int ldo,
                               const float* g, const float* b,
                               int M, int K, int N, int act)
{
    const int tiles = ((M + 15) / 16) * ((N + 31) / 32);
    dim3 blk(32, 8);
    const int grid = (tiles + 7) / 8;
    gemm_wmma_f32<<<grid, blk, 0, s>>>(A, lda, rpb, bstride, Bm, Out, ldo, g, b, M, K, N, act);
}

extern "C" void kernel_launch(void* const* d_in, const int* in_sizes, int n_in,
                              void* d_out, int out_size, void* d_ws, size_t ws_size,
                              hipStream_t stream)
{
    (void)in_sizes; (void)n_in; (void)out_size; (void)ws_size;

    int ii = 0;
    auto nxt = [&]() { return (const float*)d_in[ii++]; };

    // pytree flatten order: points, then params sorted: dense0, fc[0], fc[1], logits, xconv[0..3]
    const float* points   = nxt();
    const float* Wdense0  = nxt();
    const float* Wfc0     = nxt();
    const float* Wfc1     = nxt();
    const float* Wlogits  = nxt();

    LayerP lp[4];
    for (int l = 0; l < 4; ++l) {
        LayerP& q = lp[l];
        // 'a': W1, W2, bn1{b,g}, bn2{b,g}
        q.aW1 = nxt(); q.aW2 = nxt();
        q.abn1b = nxt(); q.abn1g = nxt(); q.abn2b = nxt(); q.abn2g = nxt();
        // 'b': Wd1, Wd2, Wp1, Wp2, Wx0, bnx0{b,g}, bnx1{b,g}, bnx2{b,g}
        q.Wd1 = nxt(); q.Wd2 = nxt(); q.Wp1 = nxt(); q.Wp2 = nxt(); q.Wx0 = nxt();
        q.bnx0b = nxt(); q.bnx0g = nxt();
        q.bnx1b = nxt(); q.bnx1g = nxt();
        q.bnx2b = nxt(); q.bnx2g = nxt();
        // cWd, cWp, cbn{b,g}
        q.cWd = nxt(); q.cWp = nxt(); q.cbnb = nxt(); q.cbng = nxt();
        q.dW1 = q.dW2 = q.dbn1b = q.dbn1g = q.dbn2b = q.dbn2g = nullptr;
        if (l == 3) {
            q.dW1 = nxt(); q.dW2 = nxt();
            q.dbn1b = nxt(); q.dbn1g = nxt(); q.dbn2b = nxt(); q.dbn2g = nxt();
        }
    }

    // workspace bump allocator
    char* ws = (char*)d_ws;
    auto alloc_f = [&](size_t nfl) {
        float* p = (float*)ws;
        ws += ((nfl * sizeof(float)) + 255) & ~(size_t)255;
        return p;
    };
    float* ftsA   = alloc_f(2097152);
    float* ftsB   = alloc_f(2097152);
    int*   idxBuf = (int*)alloc_f(262144);
    float* locBuf = alloc_f(786432);
    float* faBuf  = alloc_f(3145728);
    float* finBuf = alloc_f(15728640);
    float* Xa     = alloc_f(2097152);
    float* Xb     = alloc_f(2097152);
    float* Dw     = alloc_f(2097152);
    float* dwc    = alloc_f(4718592);
    float* dBuf   = alloc_f(393216);

    const int Kl[4]    = {8, 12, 16, 16};
    const int Dl[4]    = {1, 2, 2, 3};
    const int Pl[4]    = {1024, 384, 128, 128};
    const int Cl[4]    = {48, 96, 192, 384};
    const int cpfA[4]  = {12, 12, 24, 48};
    const int dmA[4]   = {4, 2, 2, 2};
    const int cprA[4]  = {24, 48, 96, 192};
    const int prevP[4] = {1024, 1024, 384, 128};

    // fts0 = ELU(points[...,3:] @ dense0) : M = 32768, K = 3, N = 24
    launch_gemm(stream, points + 3, 6, BB * NPTT, 0, Wdense0, ftsA, 24,
                nullptr, nullptr, BB * NPTT, 3, 24, 1);

    float* ftsPrev = ftsA;
    float* ftsOut  = ftsB;

    for (int l = 0; l < 4; ++l) {
        const LayerP& q = lp[l];
        const int K = Kl[l], D = Dl[l], P = Pl[l], C = Cl[l];
        const int cpf = cpfA[l], dm = dmA[l], cprev = cprA[l];
        const int cin = cpf + cprev;
        const int KD  = K * D;
        const int NP  = BB * P;
        const int M3  = NP * K;

        k_knn<<<(NP + 127) / 128, 128, 0, stream>>>(points, idxBuf, P, prevP[l], K, D, KD);

        k_gather<<<(M3 + 255) / 256, 256, 0, stream>>>(points, ftsPrev, idxBuf,
                                                       locBuf, finBuf, P, K, cprev, cin,
                                                       cpf, prevP[l]);

        // dense_bn2 on nn_local -> f_in[:, 0:cpf]
        launch_gemm(stream, locBuf, 3, M3, 0, q.aW1, faBuf, cpf,
                    q.abn1g, q.abn1b, M3, 3, cpf, 2);
        launch_gemm(stream, faBuf, cpf, M3, 0, q.aW2, finBuf, cin,
                    q.abn2g, q.abn2b, M3, cpf, cpf, 2);

        // X-transform
        launch_gemm(stream, locBuf, 3 * K, NP, 0, q.Wx0, Xa, K * K,
                    q.bnx0g, q.bnx0b, NP, 3 * K, K * K, 2);
        {
            const long long tot = (long long)NP * K * K;
            k_depthwise<<<(int)((tot + 255) / 256), 256, 0, stream>>>(Xa, q.Wd1, Dw, NP, K, K, K);
        }
        launch_gemm(stream, Dw, K * K, NP, 0, q.Wp1, Xb, K * K,
                    q.bnx1g, q.bnx1b, NP, K * K, K * K, 2);
        {
            const long long tot = (long long)NP * K * K;
            k_depthwise<<<(int)((tot + 255) / 256), 256, 0, stream>>>(Xb, q.Wd2, Dw, NP, K, K, K);
        }
        launch_gemm(stream, Dw, K * K, NP, 0, q.Wp2, Xa, K * K,
                    q.bnx2g, q.bnx2b, NP, K * K, K * K, 3);

        // fX = X @ f_in   (in place on f_in)
        k_fx<<<NP, 256, 0, stream>>>(Xa, finBuf, K, cin);

        // out = BN(ELU(sep_fullK(fX, cWd, cWp)))
        {
            const long long tot = (long long)NP * cin * dm;
            k_depthwise<<<(int)((tot + 255) / 256), 256, 0, stream>>>(finBuf, q.cWd, dwc,
                                                                      NP, K, cin, dm);
        }
        const int ldo = (l == 3) ? (96 + C) : C;
        const int colOff = (l == 3) ? 96 : 0;
        launch_gemm(stream, dwc, cin * dm, NP, 0, q.cWp, ftsOut + colOff, ldo,
                    q.cbng, q.cbnb, NP, cin * dm, C, 2);

        if (l == 3) {
            // dense_bn2(qrs) -> columns 0..95 ; qrs rows: batch n, point p<128
            launch_gemm(stream, points, 6, 128, (long long)NPTT * 6, q.dW1, dBuf, 96,
                        q.dbn1g, q.dbn1b, NP, 3, 96, 2);
            launch_gemm(stream, dBuf, 96, NP, 0, q.dW2, ftsOut, ldo,
                        q.dbn2g, q.dbn2b, NP, 96, 96, 2);
        }

        float* t = ftsPrev; ftsPrev = ftsOut; ftsOut = t;
    }

    // FC head: (32*128, 480) -> 384 -> 192 -> 40, then mean over points
    const int Mfc = BB * 128;
    launch_gemm(stream, ftsPrev, 480, Mfc, 0, Wfc0, ftsOut, 384,
                nullptr, nullptr, Mfc, 480, 384, 1);
    launch_gemm(stream, ftsOut, 384, Mfc, 0, Wfc1, ftsPrev, 192,
                nullptr, nullptr, Mfc, 384, 192, 1);
    launch_gemm(stream, ftsPrev, 192, Mfc, 0, Wlogits, ftsOut, 40,
                nullptr, nullptr, Mfc, 192, 40, 0);

    k_mean<<<(BB * 40 + 127) / 128, 128, 0, stream>>>(ftsOut, (float*)d_out, 128, 40);
}